// SGC_5514738008405
// MI455X (gfx1250) — compile-verified
//
#include <hip/hip_runtime.h>
#include <hip/hip_bf16.h>
#include <math.h>

// SGC on gfx1250:
//   deg/dinv prelude -> 2 hops of edge scatter (L2-resident atomics)
//   -> fused WMMA f32 head (x @ W^T + b) + log_softmax.

typedef __attribute__((ext_vector_type(2))) float v2f;
typedef __attribute__((ext_vector_type(8))) float v8f;

#define D_FEAT 128
#define C_CLS  64

// deg[i] = 1 (self loop)
__global__ void sgc_deg_init(float* __restrict__ deg, int n) {
  int i = blockIdx.x * blockDim.x + threadIdx.x;
  if (i < n) deg[i] = 1.0f;
}

// deg[dst] += 1 per directed edge
__global__ void sgc_deg_accum(const int* __restrict__ dst, float* __restrict__ deg, int e) {
  int i = blockIdx.x * blockDim.x + threadIdx.x;
  if (i < e) atomicAdd(&deg[dst[i]], 1.0f);
}

// in-place dinv = rsqrt(max(deg,1))
__global__ void sgc_dinv(float* __restrict__ deg, int n) {
  int i = blockIdx.x * blockDim.x + threadIdx.x;
  if (i < n) deg[i] = rsqrtf(fmaxf(deg[i], 1.0f));
}

// y[i,:] = dinv[i]^2 * xin[i,:]  (self-loop term; also initializes y)
// one lane per float4 (32 lanes cover the 128-feature row)
__global__ void sgc_selfloop(const float* __restrict__ dinv, const float* __restrict__ xin,
                             float* __restrict__ y, int n) {
  int t = blockIdx.x * blockDim.x + threadIdx.x;
  int i = t >> 5, j = t & 31;
  if (i >= n) return;
  float c = dinv[i] * dinv[i];
  float4 v = ((const float4*)(xin + (size_t)i * D_FEAT))[j];
  float4 r; r.x = c * v.x; r.y = c * v.y; r.z = c * v.z; r.w = c * v.w;
  ((float4*)(y + (size_t)i * D_FEAT))[j] = r;
}

// y[dst,:] += dinv[src]*dinv[dst] * xin[src,:] ; wave per edge, lane = float4 slice
__global__ void sgc_edge_prop(const int* __restrict__ src, const int* __restrict__ dst,
                              const float* __restrict__ dinv,
                              const float* __restrict__ xin, float* __restrict__ y, int e) {
  int t = blockIdx.x * blockDim.x + threadIdx.x;
  int ei = t >> 5, j = t & 31;
  if (ei >= e) return;
  int s = src[ei], d = dst[ei];
  float c = dinv[s] * dinv[d];
  float4 v = ((const float4*)(xin + (size_t)s * D_FEAT))[j];
  float* yr = y + (size_t)d * D_FEAT + (size_t)j * 4;
  atomicAdd(yr + 0, c * v.x);
  atomicAdd(yr + 1, c * v.y);
  atomicAdd(yr + 2, c * v.z);
  atomicAdd(yr + 3, c * v.w);
}

// WMMA f32 head: each wave computes a 16-row x 64-class tile, then fused
// log_softmax via LDS. Block = 128 threads = 4 waves = 64 rows.
__global__ __launch_bounds__(128)
void sgc_head(const float* __restrict__ X, const float* __restrict__ W,
              const float* __restrict__ bias, float* __restrict__ out, int n) {
  __shared__ float sm[4][16][C_CLS + 4];   // +4 pad -> conflict-free row reads

  const int wave = threadIdx.x >> 5;
  const int lane = threadIdx.x & 31;
  const int half = lane >> 4;              // 0: K=k..k+1, 1: K=k+2..k+3
  const int lid  = lane & 15;
  const int rowBase = blockIdx.x * 64 + wave * 16;
  const int m = rowBase + lid;
  const bool valid = (m < n);
  const float* xrow = X + (size_t)(valid ? m : 0) * D_FEAT;
  const float mask = valid ? 1.0f : 0.0f;  // zero A rows past N (keeps EXEC all-1)

  v8f acc0 = {}, acc1 = {}, acc2 = {}, acc3 = {};
  for (int k = 0; k < D_FEAT; k += 4) {
    const int kk = k + half * 2;
    v2f a; a.x = xrow[kk] * mask; a.y = xrow[kk + 1] * mask;     // A 16x4 f32 frag
    const float* wp = W + (size_t)lid * D_FEAT + kk;             // B[k][n] = W[n][k]
    v2f b0, b1, b2, b3;
    b0.x = wp[0];               b0.y = wp[1];
    b1.x = wp[16 * D_FEAT + 0]; b1.y = wp[16 * D_FEAT + 1];
    b2.x = wp[32 * D_FEAT + 0]; b2.y = wp[32 * D_FEAT + 1];
    b3.x = wp[48 * D_FEAT + 0]; b3.y = wp[48 * D_FEAT + 1];
    acc0 = __builtin_amdgcn_wmma_f32_16x16x4_f32(false, a, false, b0, (short)0, acc0, false, false);
    acc1 = __builtin_amdgcn_wmma_f32_16x16x4_f32(false, a, false, b1, (short)0, acc1, false, false);
    acc2 = __builtin_amdgcn_wmma_f32_16x16x4_f32(false, a, false, b2, (short)0, acc2, false, false);
    acc3 = __builtin_amdgcn_wmma_f32_16x16x4_f32(false, a, false, b3, (short)0, acc3, false, false);
  }

  // Spill D tiles (+bias) to LDS. C/D layout: VGPR v, lanes 0-15 -> row v,
  // lanes 16-31 -> row v+8, col = lane%16 within the 16-col tile.
  const float bn0 = bias[lid], bn1 = bias[16 + lid], bn2 = bias[32 + lid], bn3 = bias[48 + lid];
#pragma unroll
  for (int v = 0; v < 8; ++v) {
    const int r = v + half * 8;
    sm[wave][r][lid +  0] = acc0[v] + bn0;
    sm[wave][r][lid + 16] = acc1[v] + bn1;
    sm[wave][r][lid + 32] = acc2[v] + bn2;
    sm[wave][r][lid + 48] = acc3[v] + bn3;
  }
  __syncthreads();

  // Fused log_softmax: lanes 0-15 each own one row.
  if (lane < 16 && (rowBase + lane) < n) {
    const float* row = sm[wave][lane];
    float mx = -3.402823466e38f;
#pragma unroll 8
    for (int c = 0; c < C_CLS; ++c) mx = fmaxf(mx, row[c]);
    float sum = 0.0f;
#pragma unroll 8
    for (int c = 0; c < C_CLS; ++c) sum += __expf(row[c] - mx);
    const float lse = mx + __logf(sum);
    float* orow = out + (size_t)(rowBase + lane) * C_CLS;
#pragma unroll 8
    for (int c = 0; c < C_CLS; ++c) orow[c] = row[c] - lse;
  }
}

extern "C" void kernel_launch(void* const* d_in, const int* in_sizes, int n_in,
                              void* d_out, int out_size, void* d_ws, size_t ws_size,
                              hipStream_t stream) {
  const float* x  = (const float*)d_in[0];   // [N,128]
  const int*   ei = (const int*)d_in[1];     // [2,E] row-major
  const float* W  = (const float*)d_in[2];   // [64,128]
  const float* b  = (const float*)d_in[3];   // [64]
  float* out = (float*)d_out;                // [N,64]

  const int Nn = in_sizes[0] / D_FEAT;
  const int Ee = in_sizes[1] / 2;
  const int* src = ei;
  const int* dst = ei + Ee;

  // workspace: deg/dinv [N] (padded), y1 [N*128], y2 [N*128]  (~103 MB)
  float* deg = (float*)d_ws;
  size_t degPad = ((size_t)Nn + 255) & ~(size_t)255;
  float* y1 = deg + degPad;
  float* y2 = y1 + (size_t)Nn * D_FEAT;

  const int TB = 256;
  const int nodeBlocks = (Nn + TB - 1) / TB;
  const int edgeBlocks = (Ee + TB - 1) / TB;
  const long long nodeT = (long long)Nn * 32;
  const long long edgeT = (long long)Ee * 32;
  const int nodeVBlocks = (int)((nodeT + TB - 1) / TB);
  const int edgeVBlocks = (int)((edgeT + TB - 1) / TB);

  sgc_deg_init <<<nodeBlocks, TB, 0, stream>>>(deg, Nn);
  sgc_deg_accum<<<edgeBlocks, TB, 0, stream>>>(dst, deg, Ee);
  sgc_dinv     <<<nodeBlocks, TB, 0, stream>>>(deg, Nn);

  // hop 1: y1 = Ahat * x
  sgc_selfloop <<<nodeVBlocks, TB, 0, stream>>>(deg, x, y1, Nn);
  sgc_edge_prop<<<edgeVBlocks, TB, 0, stream>>>(src, dst, deg, x, y1, Ee);
  // hop 2: y2 = Ahat * y1
  sgc_selfloop <<<nodeVBlocks, TB, 0, stream>>>(deg, y1, y2, Nn);
  sgc_edge_prop<<<edgeVBlocks, TB, 0, stream>>>(src, dst, deg, y1, y2, Ee);

  // fused WMMA head + log_softmax
  sgc_head<<<(Nn + 63) / 64, 128, 0, stream>>>(y2, W, b, out, Nn);
}